// GCN_90563680403609
// MI455X (gfx1250) — compile-verified
//
#include <hip/hip_runtime.h>
#include <math.h>

typedef float v2f __attribute__((ext_vector_type(2)));
typedef float v8f __attribute__((ext_vector_type(8)));

#define IN_C  128
#define HID_C 16

// ---------------- degree / normalization ----------------

__global__ void k_init_deg(float* __restrict__ deg, int N) {
    int i = blockIdx.x * blockDim.x + threadIdx.x;
    if (i < N) deg[i] = 1.0f;  // self-loop contributes 1 to every node's degree
}

__global__ void k_degree(const int* __restrict__ ei, float* __restrict__ deg, int E) {
    int e = blockIdx.x * blockDim.x + threadIdx.x;
    if (e < E) {
        int d = ei[E + e];            // edge_index[1][e]
        atomicAdd(&deg[d], 1.0f);
    }
}

__global__ void k_rsqrt(float* __restrict__ deg_dinv, int N) {
    int i = blockIdx.x * blockDim.x + threadIdx.x;
    if (i < N) {
        float d = deg_dinv[i];
        deg_dinv[i] = (d > 0.0f) ? rsqrtf(d) : 0.0f;
    }
}

// ---------------- layer-1 GEMM: h1 = x @ W1 via V_WMMA_F32_16X16X4_F32 ----------------
// One wave per 16-row tile of x. 32 K-steps of 16x16x4 f32 WMMA.

__global__ void k_gemm1_wmma(const float* __restrict__ x, const float* __restrict__ W1,
                             float* __restrict__ h1, int N, int ntiles) {
    int wave = threadIdx.x >> 5;
    int tile = blockIdx.x * (blockDim.x >> 5) + wave;
    if (tile >= ntiles) return;               // uniform per wave -> EXEC all-1s inside

    int lane = threadIdx.x & 31;
    int half = lane >> 4;                     // 0: K={0,1}, 1: K={2,3}
    int l15  = lane & 15;

    int row = tile * 16 + l15;
    int rowc = row < N ? row : N - 1;         // clamp for generic N (N here is 16-aligned)
    const float* xrow = x + (size_t)rowc * IN_C + 2 * half;

    v8f acc = {};
#pragma unroll
    for (int k0 = 0; k0 < IN_C; k0 += 4) {
        // A fragment: x[row][k0 + 2*half + {0,1}] -- contiguous, 8B-aligned
        v2f a = *(const v2f*)(xrow + k0);
        // B fragment: W1[k0 + 2*half + {0,1}][l15], row-major [128][16]
        v2f b;
        b.x = W1[(k0 + 2 * half + 0) * HID_C + l15];
        b.y = W1[(k0 + 2 * half + 1) * HID_C + l15];
        acc = __builtin_amdgcn_wmma_f32_16x16x4_f32(
            /*neg_a=*/false, a, /*neg_b=*/false, b,
            /*c_mod=*/(short)0, acc, /*reuse_a=*/false, /*reuse_b=*/false);
    }

    // D layout: VGPR v -> row (tile*16 + v + 8*half), col = l15
#pragma unroll
    for (int v = 0; v < 8; ++v) {
        int r = tile * 16 + v + 8 * half;
        if (r < N) h1[(size_t)r * HID_C + l15] = acc[v];
    }
}

// ---------------- layer-1 aggregation ----------------

// agg1[i] = h1[i] * dinv[i]^2   (self-loop term; also zero-initializes agg1)
__global__ void k_self1(const float* __restrict__ h1, const float* __restrict__ dinv,
                        float* __restrict__ agg1, int N) {
    int idx = blockIdx.x * blockDim.x + threadIdx.x;   // over N*16 elements
    if (idx < N * HID_C) {
        float dv = dinv[idx >> 4];
        agg1[idx] = h1[idx] * dv * dv;
    }
}

__global__ void k_edge1(const int* __restrict__ ei, const float* __restrict__ h1,
                        const float* __restrict__ dinv, float* __restrict__ agg1, int E) {
    int e = blockIdx.x * blockDim.x + threadIdx.x;
    if (e >= E) return;
    int s = ei[e];
    int d = ei[E + e];
    float w = dinv[s] * dinv[d];
    const float4* hs = (const float4*)(h1 + (size_t)s * HID_C);
    float* out = agg1 + (size_t)d * HID_C;
#pragma unroll
    for (int q = 0; q < 4; ++q) {
        float4 v = hs[q];
        atomicAdd(out + 4 * q + 0, v.x * w);
        atomicAdd(out + 4 * q + 1, v.y * w);
        atomicAdd(out + 4 * q + 2, v.z * w);
        atomicAdd(out + 4 * q + 3, v.w * w);
    }
}

// ---------------- relu + layer-2 linear (16 -> 1), plus agg2 self-loop init ----------------

__global__ void k_mlp(const float* __restrict__ agg1, const float* __restrict__ b1,
                      const float* __restrict__ W2, const float* __restrict__ dinv,
                      float* __restrict__ h2, float* __restrict__ agg2, int N) {
    int i = blockIdx.x * blockDim.x + threadIdx.x;
    if (i >= N) return;
    const float* r = agg1 + (size_t)i * HID_C;
    float acc = 0.0f;
#pragma unroll
    for (int c = 0; c < HID_C; ++c) {
        float t = r[c] + b1[c];
        t = t > 0.0f ? t : 0.0f;               // relu
        acc += t * W2[c];                      // [16]x[16,1] dot
    }
    h2[i] = acc;
    float dv = dinv[i];
    agg2[i] = acc * dv * dv;                   // self-loop term + init
}

__global__ void k_edge2(const int* __restrict__ ei, const float* __restrict__ h2,
                        const float* __restrict__ dinv, float* __restrict__ agg2, int E) {
    int e = blockIdx.x * blockDim.x + threadIdx.x;
    if (e >= E) return;
    int s = ei[e];
    int d = ei[E + e];
    atomicAdd(&agg2[d], h2[s] * dinv[s] * dinv[d]);
}

__global__ void k_out(const float* __restrict__ agg2, const float* __restrict__ b2,
                      float* __restrict__ out, int N) {
    int i = blockIdx.x * blockDim.x + threadIdx.x;
    if (i < N) {
        float z = agg2[i] + b2[0];
        out[i] = 1.0f / (1.0f + expf(-z));     // sigmoid
    }
}

// ---------------- launch ----------------

extern "C" void kernel_launch(void* const* d_in, const int* in_sizes, int n_in,
                              void* d_out, int out_size, void* d_ws, size_t ws_size,
                              hipStream_t stream) {
    const float* x  = (const float*)d_in[0];
    const int*   ei = (const int*)d_in[1];
    const float* W1 = (const float*)d_in[2];
    const float* b1 = (const float*)d_in[3];
    const float* W2 = (const float*)d_in[4];
    const float* b2 = (const float*)d_in[5];

    const int N = in_sizes[0] / IN_C;
    const int E = in_sizes[1] / 2;

    // workspace layout (floats): dinv[N] | h1[16N] | agg1[16N] | h2[N] | agg2[N]
    float* ws   = (float*)d_ws;
    float* dinv = ws;
    float* h1   = dinv + (size_t)N;
    float* agg1 = h1   + (size_t)N * HID_C;
    float* h2   = agg1 + (size_t)N * HID_C;
    float* agg2 = h2   + (size_t)N;

    float* out = (float*)d_out;

    const int T  = 256;
    const int gN  = (N + T - 1) / T;
    const int gNC = (N * HID_C + T - 1) / T;
    const int gE  = (E + T - 1) / T;
    const int ntiles = (N + 15) / 16;
    const int wavesPerBlk = T / 32;
    const int gT = (ntiles + wavesPerBlk - 1) / wavesPerBlk;

    k_init_deg<<<gN, T, 0, stream>>>(dinv, N);
    k_degree<<<gE, T, 0, stream>>>(ei, dinv, E);
    k_rsqrt<<<gN, T, 0, stream>>>(dinv, N);

    k_gemm1_wmma<<<gT, T, 0, stream>>>(x, W1, h1, N, ntiles);

    k_self1<<<gNC, T, 0, stream>>>(h1, dinv, agg1, N);
    k_edge1<<<gE, T, 0, stream>>>(ei, h1, dinv, agg1, E);

    k_mlp<<<gN, T, 0, stream>>>(agg1, b1, W2, dinv, h2, agg2, N);
    k_edge2<<<gE, T, 0, stream>>>(ei, h2, dinv, agg2, E);
    k_out<<<gN, T, 0, stream>>>(agg2, b2, out, N);
}